// LinkPredictionGNN_18442589569957
// MI455X (gfx1250) — compile-verified
//
#include <hip/hip_runtime.h>
#include <hip/hip_bf16.h>
#include <math.h>

typedef __attribute__((ext_vector_type(2))) float v2f;
typedef __attribute__((ext_vector_type(8))) float v8f;

#define GNN_N_NODES 100000
#define GNN_N_EDGES 1600000

// ---------------------------------------------------------------------------
// zero fill (grid-stride)
// ---------------------------------------------------------------------------
__global__ void gnn_zero_f32(float* __restrict__ p, long long n) {
  long long i = (long long)blockIdx.x * blockDim.x + threadIdx.x;
  long long stride = (long long)gridDim.x * blockDim.x;
  for (; i < n; i += stride) p[i] = 0.0f;
}

// ---------------------------------------------------------------------------
// Edge scatter-add: thread = (edge, 4-feature chunk). float4 gather from
// x[src], 4 global fp32 atomics into agg[dst]; agg fits in L2 (25.6MB << 192MB)
// so the atomics resolve on-chip. Optionally accumulates degree (layer 1 only).
// ---------------------------------------------------------------------------
template <int D, bool ADD_DEG>
__global__ void gnn_scatter_add(const float* __restrict__ x,
                                const int* __restrict__ src,
                                const int* __restrict__ dst,
                                float* __restrict__ agg,
                                float* __restrict__ deg,
                                int nEdges) {
  constexpr int CH = D / 4;  // 4-float chunks per edge
  long long t = (long long)blockIdx.x * blockDim.x + threadIdx.x;
  if (t >= (long long)nEdges * CH) return;
  int e = (int)(t / CH);
  int c = (int)(t % CH) * 4;
  int s = src[e];
  int d = dst[e];
  const float4 v = *(const float4*)(x + (long long)s * D + c);
  float* a = agg + (long long)d * D + c;
  atomicAdd(a + 0, v.x);
  atomicAdd(a + 1, v.y);
  atomicAdd(a + 2, v.z);
  atomicAdd(a + 3, v.w);
  if (ADD_DEG && c == 0) atomicAdd(deg + d, 1.0f);
}

// ---------------------------------------------------------------------------
// (agg/deg) @ W + b, ReLU — fp32 WMMA 16x16x4.
// Block = 32 * (DOUT/16) threads; wave w owns output columns [16w,16w+16).
// A tile (16 nodes x 4 K): lanes 0-15 hold K=k0,k0+1; lanes 16-31 K=k0+2,k0+3.
// B tile (4 K x 16 cols): mirrored layout. C/D: vgpr j holds rows j / j+8.
// ---------------------------------------------------------------------------
template <int DIN, int DOUT>
__global__ void gnn_gemm_relu_wmma(const float* __restrict__ agg,
                                   const float* __restrict__ deg,
                                   const float* __restrict__ W,     // [DIN, DOUT] row-major
                                   const float* __restrict__ bias,  // [DOUT]
                                   float* __restrict__ out) {       // [N, DOUT]
  const int wave = threadIdx.x >> 5;
  const int lane = threadIdx.x & 31;
  const int half = lane >> 4;   // 0: K pair {0,1}; 1: K pair {2,3}
  const int lid  = lane & 15;   // A row / B column within tile
  const int node0 = blockIdx.x * 16;
  const int col = wave * 16 + lid;

  const float rinv = 1.0f / fmaxf(deg[node0 + lid], 1.0f);  // mean-aggregate
  const float* __restrict__ arow = agg + (long long)(node0 + lid) * DIN;

  v8f acc = {};
#pragma unroll
  for (int kb = 0; kb < DIN; kb += 4) {
    const int k0 = kb + 2 * half;
    v2f a, b;
    a.x = arow[k0] * rinv;
    a.y = arow[k0 + 1] * rinv;
    b.x = W[(long long)k0 * DOUT + col];
    b.y = W[(long long)(k0 + 1) * DOUT + col];
    acc = __builtin_amdgcn_wmma_f32_16x16x4_f32(
        /*neg_a=*/false, a, /*neg_b=*/false, b,
        /*c_mod=*/(short)0, acc, /*reuse_a=*/false, /*reuse_b=*/false);
  }

  const float bv = bias[col];
#pragma unroll
  for (int j = 0; j < 8; ++j) {
    const int row = node0 + j + 8 * half;  // C/D layout: vgpr j -> rows j, j+8
    out[(long long)row * DOUT + col] = fmaxf(acc[j] + bv, 0.0f);
  }
}

// ---------------------------------------------------------------------------
// head: out[n] = sigmoid(mean(h2[n,:32]) * Wd + bd)
// ---------------------------------------------------------------------------
__global__ void gnn_head(const float* __restrict__ h2,
                         const float* __restrict__ Wd,
                         const float* __restrict__ bd,
                         float* __restrict__ out, int n) {
  int i = blockIdx.x * blockDim.x + threadIdx.x;
  if (i >= n) return;
  const float4* r = (const float4*)(h2 + (long long)i * 32);
  float s = 0.0f;
#pragma unroll
  for (int j = 0; j < 8; ++j) {
    float4 v = r[j];
    s += v.x + v.y + v.z + v.w;
  }
  float z = (s * (1.0f / 32.0f)) * Wd[0] + bd[0];
  out[i] = 1.0f / (1.0f + __expf(-z));
}

// ---------------------------------------------------------------------------
// launcher
// ---------------------------------------------------------------------------
extern "C" void kernel_launch(void* const* d_in, const int* in_sizes, int n_in,
                              void* d_out, int out_size, void* d_ws, size_t ws_size,
                              hipStream_t stream) {
  (void)in_sizes; (void)n_in; (void)out_size; (void)ws_size;

  const float* x  = (const float*)d_in[0];   // [N, 64]
  const int*   es = (const int*)d_in[1];     // [E]
  const int*   ed = (const int*)d_in[2];     // [E]
  const float* W1 = (const float*)d_in[3];   // [64, 64]
  const float* b1 = (const float*)d_in[4];   // [64]
  const float* W2 = (const float*)d_in[5];   // [64, 32]
  const float* b2 = (const float*)d_in[6];   // [32]
  const float* Wd = (const float*)d_in[7];   // [1,1]
  const float* bd = (const float*)d_in[8];   // [1]
  float* out = (float*)d_out;                // [N, 1]

  const long long N = GNN_N_NODES;
  float* ws  = (float*)d_ws;
  float* deg = ws;             // N
  float* agg = deg + N;        // N*64 (reused by both layers)
  float* h1  = agg + N * 64;   // N*64
  float* h2  = h1 + N * 64;    // N*32

  // layer 1: zero deg+agg (contiguous), scatter, WMMA GEMM 64->64
  gnn_zero_f32<<<2048, 256, 0, stream>>>(ws, N * 65);
  {
    long long threads = (long long)GNN_N_EDGES * 16;  // 64/4 chunks per edge
    int blocks = (int)((threads + 255) / 256);
    gnn_scatter_add<64, true><<<blocks, 256, 0, stream>>>(x, es, ed, agg, deg, GNN_N_EDGES);
  }
  gnn_gemm_relu_wmma<64, 64><<<GNN_N_NODES / 16, 128, 0, stream>>>(agg, deg, W1, b1, h1);

  // layer 2: re-zero agg, scatter h1, WMMA GEMM 64->32
  gnn_zero_f32<<<2048, 256, 0, stream>>>(agg, N * 64);
  {
    long long threads = (long long)GNN_N_EDGES * 16;
    int blocks = (int)((threads + 255) / 256);
    gnn_scatter_add<64, false><<<blocks, 256, 0, stream>>>(h1, es, ed, agg, deg, GNN_N_EDGES);
  }
  gnn_gemm_relu_wmma<64, 32><<<GNN_N_NODES / 16, 64, 0, stream>>>(agg, deg, W2, b2, h2);

  // head: mean over features + sigmoid
  gnn_head<<<(GNN_N_NODES + 255) / 256, 256, 0, stream>>>(h2, Wd, bd, out, GNN_N_NODES);
}